// GHMLoss_5317169513087
// MI455X (gfx1250) — compile-verified
//
#include <hip/hip_runtime.h>

typedef float v4f __attribute__((ext_vector_type(4)));

#define BINSP1 11          // 10 real bins + overflow bin (g in (0.9, 1+1e-6])
#define TPB    256
#define NWAVES (TPB / 32)
#define ROWB   80          // LDS bytes per row: 64 data + 16 pad (16B aligned)

// Issue 4 async b128 loads: one 64B logit row -> this thread's LDS slot.
// Per the ISA, the instruction offset is added to both the LDS and global address,
// so the same offset:imm advances both the global and LDS side.
__device__ __forceinline__ void async_load_row(unsigned lds_addr, unsigned gbyte,
                                               const float* base) {
    asm volatile(
        "global_load_async_to_lds_b128 %0, %1, %2 offset:0 th:TH_LOAD_NT\n\t"
        "global_load_async_to_lds_b128 %0, %1, %2 offset:16 th:TH_LOAD_NT\n\t"
        "global_load_async_to_lds_b128 %0, %1, %2 offset:32 th:TH_LOAD_NT\n\t"
        "global_load_async_to_lds_b128 %0, %1, %2 offset:48 th:TH_LOAD_NT"
        :: "v"(lds_addr), "v"(gbyte), "s"((unsigned long long)(size_t)base)
        : "memory");
}

// Pass 1: async-stream logits through LDS (double buffered per thread),
// per-wave LDS histograms (exact u32 counts, f64 ce-sums), one deterministic
// partial record per block into d_ws.
__global__ __launch_bounds__(TPB, 1) void ghm_pass1(
    const float* __restrict__ logits,
    const long long* __restrict__ labels,
    int N,
    unsigned int* __restrict__ cnt_part,   // [gridDim.x][BINSP1]
    double* __restrict__ ce_part)          // [gridDim.x][BINSP1]
{
    __shared__ __align__(16) unsigned char lds_rows[2][TPB * ROWB];
    __shared__ unsigned int s_cnt[NWAVES][BINSP1];
    __shared__ double       s_ce [NWAVES][BINSP1];

    const int tid  = threadIdx.x;
    const int wv   = tid >> 5;
    const int lane = tid & 31;
    if (lane < BINSP1) { s_cnt[wv][lane] = 0u; s_ce[wv][lane] = 0.0; }
    __syncthreads();

    // Low 32 bits of a generic LDS pointer == LDS byte offset (aperture layout).
    const unsigned slot0 = (unsigned)(unsigned long long)(&lds_rows[0][0]) + (unsigned)(tid * ROWB);
    const unsigned slot1 = slot0 + (unsigned)(TPB * ROWB);

    const float L2E = 1.44269504088896340736f;   // log2(e)
    const float LN2 = 0.69314718055994530942f;   // ln(2)

    const int stride = gridDim.x * blockDim.x;
    int row = blockIdx.x * blockDim.x + tid;

    if (row < N) async_load_row(slot0, (unsigned)row * 64u, logits);

    int buf = 0;
    for (; row < N; row += stride) {
        long long lab = __builtin_nontemporal_load(labels + row);

        const int nrow = row + stride;
        if (nrow < N) {
            // WAR: make sure prior ds reads of the slot we are about to
            // overwrite have fully retired before the async engine writes it.
            asm volatile("s_wait_dscnt 0x0" ::: "memory");
            async_load_row(buf ? slot0 : slot1, (unsigned)nrow * 64u, logits);
            // 8 outstanding -> wait to <=4: current row's 4 chunks landed.
            asm volatile("s_wait_asynccnt 0x4" ::: "memory");
        } else {
            asm volatile("s_wait_asynccnt 0x0" ::: "memory");
        }

        const v4f* rp = (const v4f*)(&lds_rows[buf][tid * ROWB]);
        v4f va = rp[0];
        v4f vb = rp[1];
        v4f vc = rp[2];
        v4f vd = rp[3];

        if (lab >= 0) {   // invalid rows contribute nothing anywhere
            float x0 = va.x, x1 = va.y, x2 = va.z, x3 = va.w;
            float x4 = vb.x, x5 = vb.y, x6 = vb.z, x7 = vb.w;
            float x8 = vc.x, x9 = vc.y, x10 = vc.z, x11 = vc.w;
            float x12 = vd.x, x13 = vd.y, x14 = vd.z, x15 = vd.w;

            // rowwise max (tree)
            float m0 = fmaxf(x0, x1),   m1 = fmaxf(x2, x3);
            float m2 = fmaxf(x4, x5),   m3 = fmaxf(x6, x7);
            float m4 = fmaxf(x8, x9),   m5 = fmaxf(x10, x11);
            float m6 = fmaxf(x12, x13), m7 = fmaxf(x14, x15);
            float n0 = fmaxf(m0, m1), n1 = fmaxf(m2, m3);
            float n2 = fmaxf(m4, m5), n3 = fmaxf(m6, m7);
            float m  = fmaxf(fmaxf(n0, n1), fmaxf(n2, n3));

            // sum of exp (native v_exp_f32 via exp2f)
            float e0 = exp2f((x0 - m) * L2E),   e1 = exp2f((x1 - m) * L2E);
            float e2 = exp2f((x2 - m) * L2E),   e3 = exp2f((x3 - m) * L2E);
            float e4 = exp2f((x4 - m) * L2E),   e5 = exp2f((x5 - m) * L2E);
            float e6 = exp2f((x6 - m) * L2E),   e7 = exp2f((x7 - m) * L2E);
            float e8 = exp2f((x8 - m) * L2E),   e9 = exp2f((x9 - m) * L2E);
            float e10 = exp2f((x10 - m) * L2E), e11 = exp2f((x11 - m) * L2E);
            float e12 = exp2f((x12 - m) * L2E), e13 = exp2f((x13 - m) * L2E);
            float e14 = exp2f((x14 - m) * L2E), e15 = exp2f((x15 - m) * L2E);
            float s = (((e0 + e1) + (e2 + e3)) + ((e4 + e5) + (e6 + e7)))
                    + (((e8 + e9) + (e10 + e11)) + ((e12 + e13) + (e14 + e15)));

            // select x[label] without dynamic indexing (stays in VGPRs)
            float xl = x0;
            xl = (lab == 1)  ? x1  : xl;  xl = (lab == 2)  ? x2  : xl;
            xl = (lab == 3)  ? x3  : xl;  xl = (lab == 4)  ? x4  : xl;
            xl = (lab == 5)  ? x5  : xl;  xl = (lab == 6)  ? x6  : xl;
            xl = (lab == 7)  ? x7  : xl;  xl = (lab == 8)  ? x8  : xl;
            xl = (lab == 9)  ? x9  : xl;  xl = (lab == 10) ? x10 : xl;
            xl = (lab == 11) ? x11 : xl;  xl = (lab == 12) ? x12 : xl;
            xl = (lab == 13) ? x13 : xl;  xl = (lab == 14) ? x14 : xl;
            xl = (lab == 15) ? x15 : xl;

            float t  = (xl - m) * L2E;
            float pl = exp2f(t) / s;            // correct-class probability
            float g  = 1.0f - pl;
            float ce = LN2 * (log2f(s) - t);    // -log_softmax[label]

            // searchsorted(edges, g, 'left') == #edges strictly below g.
            // Edges match f32 arange(11)/10, last edge + 1e-6f, exactly.
            int bin = 0;
            bin += (g > 0.0f); bin += (g > 0.1f); bin += (g > 0.2f);
            bin += (g > 0.3f); bin += (g > 0.4f); bin += (g > 0.5f);
            bin += (g > 0.6f); bin += (g > 0.7f); bin += (g > 0.8f);
            bin += (g > 0.9f); bin += (g > (1.0f + 1e-6f));
            if (bin > BINSP1 - 1) bin = BINSP1 - 1;   // safety clamp

            atomicAdd(&s_cnt[wv][bin], 1u);           // LDS ds_add
            atomicAdd(&s_ce[wv][bin], (double)ce);    // LDS f64 atomic
        }
        buf ^= 1;
    }
    __syncthreads();

    if (tid < BINSP1) {
        unsigned int ct = 0; double cs = 0.0;
        #pragma unroll
        for (int w = 0; w < NWAVES; ++w) { ct += s_cnt[w][tid]; cs += s_ce[w][tid]; }
        cnt_part[(size_t)blockIdx.x * BINSP1 + tid] = ct;
        ce_part [(size_t)blockIdx.x * BINSP1 + tid] = cs;
    }
}

// Pass 2: deterministic reduction of block partials + GHM finalize -> scalar.
__global__ __launch_bounds__(TPB, 1) void ghm_finalize(
    const unsigned int* __restrict__ cnt_part,
    const double* __restrict__ ce_part,
    const float* __restrict__ acc_sum,   // [10]
    int nblocks, int N,
    float* __restrict__ out)
{
    __shared__ double             redd[TPB];
    __shared__ unsigned long long redc[TPB];
    __shared__ double             ce_sum_s[BINSP1];
    __shared__ unsigned long long cnt_s[BINSP1];

    const int tid = threadIdx.x;
    for (int b = 0; b < BINSP1; ++b) {
        double acc = 0.0; unsigned long long cc = 0ull;
        for (int i = tid; i < nblocks; i += TPB) {
            acc += ce_part[(size_t)i * BINSP1 + b];
            cc  += cnt_part[(size_t)i * BINSP1 + b];
        }
        redd[tid] = acc; redc[tid] = cc;
        __syncthreads();
        for (int off = TPB / 2; off > 0; off >>= 1) {
            if (tid < off) { redd[tid] += redd[tid + off]; redc[tid] += redc[tid + off]; }
            __syncthreads();
        }
        if (tid == 0) { ce_sum_s[b] = redd[0]; cnt_s[b] = redc[0]; }
        __syncthreads();
    }

    if (tid == 0) {
        double tv = 0.0;
        for (int b = 0; b < BINSP1; ++b) tv += (double)cnt_s[b];
        float total_valid = (float)tv;        // N <= 2^24 -> exact in f32

        double sum_raw = 0.0, S = 0.0;
        for (int b = 0; b < BINSP1 - 1; ++b) {   // overflow bin has weight 0
            float w = 0.0f;
            float cb = (float)cnt_s[b];
            if (cnt_s[b] > 0ull) {
                float na = 0.75f * acc_sum[b] + 0.25f * cb;  // EMA update
                na = fmaxf(na, 1e-12f);
                w = total_valid / na;
            }
            sum_raw += (double)w * (double)cb;
            S       += (double)w * ce_sum_s[b];
        }
        double res = S * (double)total_valid / (sum_raw * (double)N);
        out[0] = (float)res;
    }
}

extern "C" void kernel_launch(void* const* d_in, const int* in_sizes, int n_in,
                              void* d_out, int out_size, void* d_ws, size_t ws_size,
                              hipStream_t stream) {
    const float*     logits  = (const float*)d_in[0];
    const float*     acc_sum = (const float*)d_in[1];
    const long long* labels  = (const long long*)d_in[2];
    float*           out     = (float*)d_out;
    const int N = in_sizes[2];          // labels element count == rows

    int maxData = (N + TPB - 1) / TPB; if (maxData < 1) maxData = 1;
    int gblocks = 2048;                 // ~16K waves; ~8 rows/thread pipeline depth
    if (gblocks > maxData) gblocks = maxData;
    if (ws_size >= 1024) {
        size_t cap = (ws_size - 512) / (size_t)(BINSP1 * 12 + 16);
        if ((size_t)gblocks > cap) gblocks = (int)cap;
    } else {
        gblocks = 1;
    }
    if (gblocks < 1) gblocks = 1;

    unsigned int* cnt_part = (unsigned int*)d_ws;
    size_t cntBytes = (((size_t)gblocks * BINSP1 * sizeof(unsigned int)) + 255u) & ~(size_t)255u;
    double* ce_part = (double*)((char*)d_ws + cntBytes);

    ghm_pass1<<<gblocks, TPB, 0, stream>>>(logits, labels, N, cnt_part, ce_part);
    ghm_finalize<<<1, TPB, 0, stream>>>(cnt_part, ce_part, acc_sum, gblocks, N, out);
}